// DualRelationModule_19713899889183
// MI455X (gfx1250) — compile-verified
//
#include <hip/hip_runtime.h>

#define HWD   12288
#define WID   96
#define HGT   128
#define PN    64
#define CCH   32
#define KCH   17
#define GN    16
#define NI    4
#define CK    49
#define WPAD  52      // padded input-channel pitch for conv weights
#define MPAD  20      // padded input-channel pitch for folded attention matrices

typedef float v2f __attribute__((ext_vector_type(2)));
typedef float v8f __attribute__((ext_vector_type(8)));

__device__ __forceinline__ v8f wmma4(v2f a, v2f b, v8f c) {
  // D = A(16x4 f32) * B(4x16 f32) + C(16x16 f32)
  return __builtin_amdgcn_wmma_f32_16x16x4_f32(false, a, false, b, (short)0, c, false, false);
}

// ---------------------------------------------------------------------------
// Copy conv weights/bias into zero-padded [32][WPAD] layout (region pre-zeroed).
// ---------------------------------------------------------------------------
__global__ void k_pad_w(const float* __restrict__ w, const float* __restrict__ b,
                        int Cout, float* __restrict__ wp, float* __restrict__ bp) {
  int t = blockIdx.x*256 + threadIdx.x;
  if (t < Cout*CK) { int o = t / CK, c = t % CK; wp[o*WPAD + c] = w[t]; }
  if (t < Cout) bp[t] = b[t];
}

// ---------------------------------------------------------------------------
// Gram over spatial dim, per (channel, group): G[c,g,n,m] = sum_d x[n,c,d]x[m,c,d]
// Also per-row sums for SE means. grid.x = CCH*GN*3, block 256, chunk 4096
// ---------------------------------------------------------------------------
__global__ void k_gram_group(const float* __restrict__ x, float* __restrict__ Gout,
                             float* __restrict__ sumOut) {
  const int b  = blockIdx.x;
  const int c  = b % CCH;
  const int g  = (b / CCH) % GN;
  const int ch = b / (CCH * GN);
  const int d0 = ch * 4096;
  const float* x0 = x + ((size_t)((g*NI+0)*CCH + c))*HWD + d0;
  const float* x1 = x + ((size_t)((g*NI+1)*CCH + c))*HWD + d0;
  const float* x2 = x + ((size_t)((g*NI+2)*CCH + c))*HWD + d0;
  const float* x3 = x + ((size_t)((g*NI+3)*CCH + c))*HWD + d0;
  float pr[10] = {0,0,0,0,0,0,0,0,0,0};
  float rs[4]  = {0,0,0,0};
  for (int i = threadIdx.x; i < 4096; i += 256) {
    float v0 = x0[i], v1 = x1[i], v2 = x2[i], v3 = x3[i];
    rs[0]+=v0; rs[1]+=v1; rs[2]+=v2; rs[3]+=v3;
    pr[0]+=v0*v0; pr[1]+=v0*v1; pr[2]+=v0*v2; pr[3]+=v0*v3;
    pr[4]+=v1*v1; pr[5]+=v1*v2; pr[6]+=v1*v3;
    pr[7]+=v2*v2; pr[8]+=v2*v3; pr[9]+=v3*v3;
  }
  __shared__ float red[14];
  if (threadIdx.x < 14) red[threadIdx.x] = 0.f;
  __syncthreads();
  for (int j = 0; j < 10; j++) atomicAdd(&red[j], pr[j]);
  for (int n = 0; n < 4;  n++) atomicAdd(&red[10+n], rs[n]);
  __syncthreads();
  if (threadIdx.x < 14) {
    const int pnA[10] = {0,0,0,0,1,1,1,2,2,3};
    const int pmA[10] = {0,1,2,3,1,2,3,2,3,3};
    int j = threadIdx.x;
    if (j < 10) {
      int n = pnA[j], m = pmA[j];
      float v = red[j];
      atomicAdd(&Gout[((size_t)c*GN + g)*16 + n*4 + m], v);
      if (n != m) atomicAdd(&Gout[((size_t)c*GN + g)*16 + m*4 + n], v);
    } else if (sumOut) {
      int n = j - 10;
      atomicAdd(&sumOut[(g*NI + n)*CCH + c], red[j]);
    }
  }
}

// ---------------------------------------------------------------------------
// Per-instance 17x17 Gram over spatial dim + row sums (for joint_rel folding).
// grid (PN, HWD/768), block 256. LDS tile [17][768].
// ---------------------------------------------------------------------------
__global__ void k_gram_self(const float* __restrict__ x, float* __restrict__ Gout,
                            float* __restrict__ Sout) {
  __shared__ float t[KCH * 768];
  const int p  = blockIdx.x;
  const int d0 = blockIdx.y * 768;
  for (int i = threadIdx.x; i < KCH * 768; i += 256) {
    int k = i / 768, d = i % 768;
    t[i] = x[((size_t)p*KCH + k)*HWD + d0 + d];
  }
  __syncthreads();
  const int tid = threadIdx.x;
  if (tid < 153) {
    int k = 0, rem = tid;
    while (rem >= (KCH - k)) { rem -= (KCH - k); k++; }
    int l = k + rem;
    const float* a  = &t[k*768];
    const float* bb = &t[l*768];
    float acc = 0.f;
    for (int i = 0; i < 768; i++) acc += a[i]*bb[i];
    atomicAdd(&Gout[(size_t)p*289 + k*KCH + l], acc);
    if (k != l) atomicAdd(&Gout[(size_t)p*289 + l*KCH + k], acc);
  }
  if (tid >= 160 && tid < 160 + KCH) {
    int k = tid - 160;
    const float* a = &t[k*768];
    float acc = 0.f;
    for (int i = 0; i < 768; i++) acc += a[i];
    atomicAdd(&Sout[p*KCH + k], acc);
  }
}

// ---------------------------------------------------------------------------
// inst_rel attention: att[c,g,n,m] = softmax_c(G/96 + paramGram/480). 1 block.
// ---------------------------------------------------------------------------
__global__ void k_att_group(const float* __restrict__ Gin, const float* __restrict__ param,
                            float* __restrict__ attOut) {
  const int t = threadIdx.x;
  const int g = t >> 4, n = (t >> 2) & 3, m = t & 3;
  const float* pn_ = param + (g*NI + n)*34;
  const float* pm_ = param + (g*NI + m)*34;
  float pg = 0.f;
  for (int d = 0; d < 34; d++) pg += pn_[d]*pm_[d];
  pg *= (1.0f/480.0f);
  const size_t idx0 = (size_t)g*16 + n*4 + m;
  float mx = -1e30f;
  for (int c = 0; c < CCH; c++) {
    float v = Gin[(size_t)c*GN*16 + idx0]*(1.f/96.f) + pg;
    mx = fmaxf(mx, v);
  }
  float s = 0.f;
  for (int c = 0; c < CCH; c++)
    s += expf(Gin[(size_t)c*GN*16 + idx0]*(1.f/96.f) + pg - mx);
  for (int c = 0; c < CCH; c++) {
    float e = expf(Gin[(size_t)c*GN*16 + idx0]*(1.f/96.f) + pg - mx);
    attOut[(size_t)c*GN*16 + idx0] = e / s;
  }
}

// ---------------------------------------------------------------------------
// joint_rel folding: rel = (Wq G Wk^T + bias terms)/96 ; softmax over instance n;
// M = att @ Wv (padded [32][MPAD]), c = att @ bv (padded [32]). grid 16, block 320.
// qkv weights: [0]=K-proj, [1]=Q-proj, [2]=V-proj. Mout/cOut pre-zeroed.
// ---------------------------------------------------------------------------
__global__ void k_jr_prep(const float* __restrict__ Gs, const float* __restrict__ Ss,
                          const float* __restrict__ qw, const float* __restrict__ qb,
                          float* __restrict__ Mout, float* __restrict__ cOut) {
  __shared__ float rel[NI][289];
  const int g = blockIdx.x, t = threadIdx.x;
  const float* Wk = qw;        const float* bk = qb;
  const float* Wq = qw + 289;  const float* bq = qb + KCH;
  const float* Wv = qw + 578;  const float* bv = qb + 2*KCH;
  if (t < 289) {
    int k = t / KCH, l = t % KCH;
    for (int n = 0; n < NI; n++) {
      int p = g*NI + n;
      const float* Gp = Gs + (size_t)p*289;
      const float* Sp = Ss + p*KCH;
      float acc = 0.f, wqs = 0.f, wks = 0.f;
      for (int a = 0; a < KCH; a++) {
        float inner = 0.f;
        for (int b2 = 0; b2 < KCH; b2++) inner += Gp[a*KCH + b2]*Wk[l*KCH + b2];
        acc += Wq[k*KCH + a]*inner;
        wqs += Wq[k*KCH + a]*Sp[a];
        wks += Wk[l*KCH + a]*Sp[a];
      }
      acc += wqs*bk[l] + bq[k]*wks + bq[k]*bk[l]*(float)HWD;
      rel[n][t] = acc * (1.0f/96.0f);
    }
  }
  __syncthreads();
  if (t < 289) {
    float mx = fmaxf(fmaxf(rel[0][t], rel[1][t]), fmaxf(rel[2][t], rel[3][t]));
    float e[NI], s = 0.f;
    for (int n = 0; n < NI; n++) { e[n] = expf(rel[n][t]-mx); s += e[n]; }
    for (int n = 0; n < NI; n++) rel[n][t] = e[n]/s;
  }
  __syncthreads();
  if (t < 289) {
    int k = t / KCH, b2 = t % KCH;
    for (int n = 0; n < NI; n++) {
      float m = 0.f;
      for (int l = 0; l < KCH; l++) m += rel[n][k*KCH + l]*Wv[l*KCH + b2];
      Mout[((size_t)(g*NI + n))*(32*MPAD) + k*MPAD + b2] = m;
      if (b2 == 0) {
        float cc = 0.f;
        for (int l = 0; l < KCH; l++) cc += rel[n][k*KCH + l]*bv[l];
        cOut[(g*NI + n)*32 + k] = cc;
      }
    }
  }
}

// ---------------------------------------------------------------------------
// inst_rel apply: out[n,c,d] = relu(sum_m att[c,g,n,m] x[m,c,d] + x[n,c,d])
// + optional channel sum/max stats. grid (CCH*GN, HWD/2048), block 256.
// ---------------------------------------------------------------------------
__global__ void k_fuse_ir(const float* __restrict__ x, const float* __restrict__ att,
                          float* __restrict__ out, int outPitch, int outBase,
                          float* sums, float* maxs, int statPitch, int statBase) {
  const int cb = blockIdx.x;
  const int c  = cb % CCH;
  const int g  = cb / CCH;
  const int d0 = blockIdx.y * 2048;
  float a[NI][NI];
  for (int n = 0; n < NI; n++)
    for (int m = 0; m < NI; m++)
      a[n][m] = att[((size_t)c*GN + g)*16 + n*4 + m];
  const float* xin[NI]; float* op[NI];
  for (int n = 0; n < NI; n++) {
    xin[n] = x   + ((size_t)((g*NI + n)*CCH + c))*HWD + d0;
    op[n]  = out + ((size_t)((g*NI + n)*outPitch + outBase + c))*HWD + d0;
  }
  float sm[NI]  = {0,0,0,0};
  float mxv[NI] = {0,0,0,0};
  for (int i = threadIdx.x; i < 2048; i += 256) {
    float v0 = xin[0][i], v1 = xin[1][i], v2 = xin[2][i], v3 = xin[3][i];
    float vv[NI] = {v0, v1, v2, v3};
    for (int n = 0; n < NI; n++) {
      float r = a[n][0]*v0 + a[n][1]*v1 + a[n][2]*v2 + a[n][3]*v3 + vv[n];
      r = fmaxf(r, 0.f);
      op[n][i] = r;
      sm[n] += r;
      mxv[n] = fmaxf(mxv[n], r);
    }
  }
  if (sums || maxs) {
    __shared__ float ssum[NI]; __shared__ int smax[NI];
    if (threadIdx.x < NI) { ssum[threadIdx.x] = 0.f; smax[threadIdx.x] = 0; }
    __syncthreads();
    for (int n = 0; n < NI; n++) {
      atomicAdd(&ssum[n], sm[n]);
      atomicMax(&smax[n], __float_as_int(mxv[n]));
    }
    __syncthreads();
    if (threadIdx.x < NI) {
      int n = threadIdx.x;
      int sidx = (g*NI + n)*statPitch + statBase + c;
      if (sums) atomicAdd(&sums[sidx], ssum[n]);
      if (maxs) atomicMax((int*)&maxs[sidx], smax[n]);
    }
  }
}

// ---------------------------------------------------------------------------
// joint_rel apply (WMMA): out[k,d] = relu(M[p] @ x[:,d] + c[p] + x[k,d])
// M is padded [32][MPAD] (zeros beyond 17) -> unconditional B loads from LDS;
// A channel index clamped (zero B columns kill padded terms).
// grid (PN, HWD/128), block 256 = 8 waves x 16 pixels.
// ---------------------------------------------------------------------------
__global__ void __launch_bounds__(256) k_fuse_kr(
    const float* __restrict__ x, const float* __restrict__ Mw, const float* __restrict__ cw,
    float* __restrict__ out, int outPitch, int outBase,
    float* sums, float* maxs, int statPitch, int statBase) {
  __shared__ float Ml[32*MPAD];
  __shared__ float ssum[KCH]; __shared__ int smax[KCH];
  const int p    = blockIdx.x;
  const int wave = threadIdx.x >> 5;
  const int lane = threadIdx.x & 31;
  const int half = lane >> 4;
  const int m16  = lane & 15;
  const int dA   = blockIdx.y*128 + wave*16 + m16;
  const float* xp = x  + (size_t)p*KCH*HWD;
  const float* Mp = Mw + (size_t)p*(32*MPAD);
  #pragma unroll
  for (int i = 0; i < 32*MPAD; i += 256) {
    int idx = i + threadIdx.x;
    if (idx < 32*MPAD) Ml[idx] = Mp[idx];
  }
  if (threadIdx.x < KCH) { ssum[threadIdx.x] = 0.f; smax[threadIdx.x] = 0; }
  __syncthreads();
  v8f acc0 = {0,0,0,0,0,0,0,0};
  v8f acc1 = {0,0,0,0,0,0,0,0};
  #pragma unroll
  for (int kc = 0; kc < 5; kc++) {
    int l0  = kc*4 + half*2;
    int lc0 = (l0   < KCH) ? l0   : KCH-1;  // clamp address; padded B col is 0
    int lc1 = (l0+1 < KCH) ? l0+1 : KCH-1;
    v2f a, b0, b1;
    a.x  = xp[(size_t)lc0*HWD + dA];
    a.y  = xp[(size_t)lc1*HWD + dA];
    b0.x = Ml[m16*MPAD + l0];
    b0.y = Ml[m16*MPAD + l0 + 1];
    b1.x = Ml[(16+m16)*MPAD + l0];
    b1.y = Ml[(16+m16)*MPAD + l0 + 1];
    acc0 = wmma4(a, b0, acc0);
    acc1 = wmma4(a, b1, acc1);
  }
  const int dBase = blockIdx.y*128 + wave*16;
  const bool st = (sums != nullptr) || (maxs != nullptr);
  #pragma unroll
  for (int r = 0; r < 8; r++) {
    int dd = dBase + r + half*8;
    {
      int o = m16;
      float v = acc0[r] + cw[p*32 + o] + xp[(size_t)o*HWD + dd];
      v = fmaxf(v, 0.f);
      out[((size_t)p*outPitch + outBase + o)*HWD + dd] = v;
      if (st) { atomicAdd(&ssum[o], v); atomicMax(&smax[o], __float_as_int(v)); }
    }
    if (m16 == 0) {
      int o = 16;
      float v = acc1[r] + cw[p*32 + o] + xp[(size_t)o*HWD + dd];
      v = fmaxf(v, 0.f);
      out[((size_t)p*outPitch + outBase + o)*HWD + dd] = v;
      if (st) { atomicAdd(&ssum[o], v); atomicMax(&smax[o], __float_as_int(v)); }
    }
  }
  __syncthreads();
  if (st && threadIdx.x < KCH) {
    int o = threadIdx.x;
    int sidx = p*statPitch + statBase + o;
    if (sums) atomicAdd(&sums[sidx], ssum[o]);
    if (maxs) atomicMax((int*)&maxs[sidx], smax[o]);
  }
}

// ---------------------------------------------------------------------------
// SE scale vector. grid PN, block 64.
// ---------------------------------------------------------------------------
__global__ void k_se(const float* __restrict__ sumA, int Ca,
                     const float* __restrict__ sumB, int Cb,
                     const float* __restrict__ w1, const float* __restrict__ b1,
                     const float* __restrict__ w2, const float* __restrict__ b2,
                     float* __restrict__ sOut) {
  __shared__ float hid[3];
  const int p = blockIdx.x, t = threadIdx.x;
  if (t < 3) {
    float h = b1[t];
    for (int c = 0; c < CK; c++) {
      float v = (c < Ca ? sumA[p*Ca + c] : sumB[p*Cb + (c - Ca)]) * (1.0f/(float)HWD);
      h += v * w1[t*CK + c];
    }
    hid[t] = fmaxf(h, 0.f);
  }
  __syncthreads();
  if (t < CK) {
    float o = b2[t];
    for (int h = 0; h < 3; h++) o += hid[h]*w2[t*3 + h];
    sOut[p*CK + t] = 1.f/(1.f + expf(-o));
  }
}

// ---------------------------------------------------------------------------
// SE-scaled concat + conv1x1 (WMMA), padded weights [32][WPAD] staged in LDS.
// grid (PN, HWD/128), block 256.
// ---------------------------------------------------------------------------
__device__ __forceinline__ float ldin(const float* inA, int Ca, const float* inB, int Cb,
                                      int p, int c, int d, const float* sp) {
  float v = (c < Ca) ? inA[((size_t)p*Ca + c)*HWD + d]
                     : inB[((size_t)p*Cb + (c - Ca))*HWD + d];
  return v * sp[c];
}

__global__ void __launch_bounds__(256) k_conv(
    const float* __restrict__ inA, int Ca, const float* __restrict__ inB, int Cb,
    const float* __restrict__ s, const float* __restrict__ Wp, const float* __restrict__ bias,
    int Cout, float* __restrict__ out) {
  __shared__ float Wl[32*WPAD];
  const int p    = blockIdx.x;
  const int wave = threadIdx.x >> 5;
  const int lane = threadIdx.x & 31;
  const int half = lane >> 4;
  const int m16  = lane & 15;
  const int dA   = blockIdx.y*128 + wave*16 + m16;
  const float* sp = s + p*CK;
  #pragma unroll
  for (int i = 0; i < 32*WPAD; i += 256) {
    int idx = i + threadIdx.x;
    if (idx < 32*WPAD) Wl[idx] = Wp[idx];
  }
  __syncthreads();
  v8f acc0 = {0,0,0,0,0,0,0,0};
  v8f acc1 = {0,0,0,0,0,0,0,0};
  #pragma unroll
  for (int kc = 0; kc < 13; kc++) {
    int c0  = kc*4 + half*2;
    int cc0 = (c0   < CK) ? c0   : CK-1;   // clamp; padded B col is 0
    int cc1 = (c0+1 < CK) ? c0+1 : CK-1;
    v2f a, b0, b1;
    a.x  = ldin(inA, Ca, inB, Cb, p, cc0, dA, sp);
    a.y  = ldin(inA, Ca, inB, Cb, p, cc1, dA, sp);
    b0.x = Wl[m16*WPAD + c0];
    b0.y = Wl[m16*WPAD + c0 + 1];
    b1.x = Wl[(16+m16)*WPAD + c0];
    b1.y = Wl[(16+m16)*WPAD + c0 + 1];
    acc0 = wmma4(a, b0, acc0);
    acc1 = wmma4(a, b1, acc1);
  }
  const int dBase = blockIdx.y*128 + wave*16;
  #pragma unroll
  for (int r = 0; r < 8; r++) {
    int dd = dBase + r + half*8;
    out[((size_t)p*Cout + m16)*HWD + dd] = acc0[r] + bias[m16];
    int o1 = 16 + m16;
    if (o1 < Cout) out[((size_t)p*Cout + o1)*HWD + dd] = acc1[r] + bias[o1];
  }
}

// ---------------------------------------------------------------------------
// Channel attention. grid PN, block 64.
// ---------------------------------------------------------------------------
__global__ void k_chatt(const float* __restrict__ sum, const float* __restrict__ mx,
                        const float* __restrict__ w1, const float* __restrict__ w2,
                        float* __restrict__ caOut) {
  __shared__ float hid[2][3];
  const int p = blockIdx.x, t = threadIdx.x;
  if (t < 6) {
    int which = t / 3, h = t % 3;
    float acc = 0.f;
    for (int c = 0; c < CK; c++) {
      float v = which ? mx[p*CK + c] : sum[p*CK + c]*(1.0f/(float)HWD);
      acc += v * w1[h*CK + c];
    }
    hid[which][h] = fmaxf(acc, 0.f);
  }
  __syncthreads();
  if (t < CK) {
    float o = 0.f;
    for (int h = 0; h < 3; h++) o += (hid[0][h] + hid[1][h])*w2[t*3 + h];
    caOut[p*CK + t] = 1.f/(1.f + expf(-o));
  }
}

// ---------------------------------------------------------------------------
// Per-pixel channel mean/max of ca-scaled ff -> 2-channel spatial map.
// grid (PN, HWD/256), block 256.
// ---------------------------------------------------------------------------
__global__ void k_smap(const float* __restrict__ ff, const float* __restrict__ ca,
                       float* __restrict__ smap) {
  const int p = blockIdx.x;
  const int d = blockIdx.y*256 + threadIdx.x;
  const float* cap = ca + p*CK;
  float s = 0.f, m = -1e30f;
  for (int c = 0; c < CK; c++) {
    float v = ff[((size_t)p*CK + c)*HWD + d] * cap[c];
    s += v; m = fmaxf(m, v);
  }
  smap[((size_t)p*2 + 0)*HWD + d] = s * (1.0f/(float)CK);
  smap[((size_t)p*2 + 1)*HWD + d] = m;
}

// ---------------------------------------------------------------------------
// Final: spatial att (7x7 conv on smap, sigmoid) * ca-scaled ff -> conv1x1(49->32)
// (WMMA, padded weights in LDS) + bias + residual + relu. grid (PN, HWD/128).
// ---------------------------------------------------------------------------
__global__ void __launch_bounds__(256) k_final(
    const float* __restrict__ ff, const float* __restrict__ ca,
    const float* __restrict__ smap, const float* __restrict__ saw,
    const float* __restrict__ fwp, const float* __restrict__ fb,
    const float* __restrict__ xin, float* __restrict__ out) {
  __shared__ float Wl[32*WPAD];
  const int p    = blockIdx.x;
  const int wave = threadIdx.x >> 5;
  const int lane = threadIdx.x & 31;
  const int half = lane >> 4;
  const int m16  = lane & 15;
  const int dA   = blockIdx.y*128 + wave*16 + m16;
  #pragma unroll
  for (int i = 0; i < 32*WPAD; i += 256) {
    int idx = i + threadIdx.x;
    if (idx < 32*WPAD) Wl[idx] = fwp[idx];
  }
  __syncthreads();
  const int y = dA / WID, xpix = dA % WID;
  const float* s0 = smap + ((size_t)p*2)*HWD;
  const float* s1 = s0 + HWD;
  float sacc = 0.f;
  for (int i = 0; i < 7; i++) {
    int yy = y + i - 3;
    if (yy < 0 || yy >= HGT) continue;
    for (int j = 0; j < 7; j++) {
      int xx = xpix + j - 3;
      if (xx < 0 || xx >= WID) continue;
      int dd = yy*WID + xx;
      sacc += saw[i*7 + j]*s0[dd] + saw[49 + i*7 + j]*s1[dd];
    }
  }
  float satt = 1.f/(1.f + expf(-sacc));
  const float* cap = ca + p*CK;
  v8f acc0 = {0,0,0,0,0,0,0,0};
  v8f acc1 = {0,0,0,0,0,0,0,0};
  #pragma unroll
  for (int kc = 0; kc < 13; kc++) {
    int c0  = kc*4 + half*2;
    int cc0 = (c0   < CK) ? c0   : CK-1;
    int cc1 = (c0+1 < CK) ? c0+1 : CK-1;
    v2f a, b0, b1;
    a.x  = ff[((size_t)p*CK + cc0)*HWD + dA]*cap[cc0]*satt;
    a.y  = ff[((size_t)p*CK + cc1)*HWD + dA]*cap[cc1]*satt;
    b0.x = Wl[m16*WPAD + c0];
    b0.y = Wl[m16*WPAD + c0 + 1];
    b1.x = Wl[(16+m16)*WPAD + c0];
    b1.y = Wl[(16+m16)*WPAD + c0 + 1];
    acc0 = wmma4(a, b0, acc0);
    acc1 = wmma4(a, b1, acc1);
  }
  const int dBase = blockIdx.y*128 + wave*16;
  #pragma unroll
  for (int r = 0; r < 8; r++) {
    int dd = dBase + r + half*8;
    int o0 = m16, o1 = 16 + m16;
    float v0 = acc0[r] + fb[o0] + xin[((size_t)p*CCH + o0)*HWD + dd];
    float v1 = acc1[r] + fb[o1] + xin[((size_t)p*CCH + o1)*HWD + dd];
    out[((size_t)p*CCH + o0)*HWD + dd] = fmaxf(v0, 0.f);
    out[((size_t)p*CCH + o1)*HWD + dd] = fmaxf(v1, 0.f);
  }
}

// ---------------------------------------------------------------------------
extern "C" void kernel_launch(void* const* d_in, const int* in_sizes, int n_in,
                              void* d_out, int out_size, void* d_ws, size_t ws_size,
                              hipStream_t stream) {
  (void)in_sizes; (void)n_in; (void)out_size; (void)ws_size;
  const float* inst_f  = (const float*)d_in[0];
  const float* kpt_f   = (const float*)d_in[1];
  const float* param   = (const float*)d_in[2];
  const float* jr1_w   = (const float*)d_in[3];
  const float* jr1_b   = (const float*)d_in[4];
  const float* jr2_w   = (const float*)d_in[5];
  const float* jr2_b   = (const float*)d_in[6];
  const float* se1_w1  = (const float*)d_in[7];
  const float* se1_b1  = (const float*)d_in[8];
  const float* se1_w2  = (const float*)d_in[9];
  const float* se1_b2  = (const float*)d_in[10];
  const float* se2_w1  = (const float*)d_in[11];
  const float* se2_b1  = (const float*)d_in[12];
  const float* se2_w2  = (const float*)d_in[13];
  const float* se2_b2  = (const float*)d_in[14];
  const float* ca_w1   = (const float*)d_in[15];
  const float* ca_w2   = (const float*)d_in[16];
  const float* sa_w    = (const float*)d_in[17];
  const float* ikc_w   = (const float*)d_in[18];
  const float* ikc_b   = (const float*)d_in[19];
  const float* kic_w   = (const float*)d_in[20];
  const float* kic_b   = (const float*)d_in[21];
  const float* ff_w    = (const float*)d_in[22];
  const float* ff_b    = (const float*)d_in[23];

  float* ws = (float*)d_ws;
  const size_t NIR1 = (size_t)PN*CCH*HWD;   // 25165824
  const size_t NKR  = (size_t)PN*KCH*HWD;   // 13369344
  float* ir1  = ws;                // [p][32][hw]
  float* kr   = ws + NIR1;         // [p][17][hw]
  float* ff   = ws;                // alias over ir1+kr: [p][49][hw] (exact size match)
  float* ikc  = ws + NIR1 + NKR;   // [p][17][hw]
  float* kic  = ikc + NKR;         // [p][32][hw]
  float* smap = kic + NIR1;        // [p][2][hw]
  float* sm   = smap + (size_t)PN*2*HWD;
  size_t o = 0;
  float* G_inst  = sm + o; o += (size_t)CCH*GN*16;   // 8192
  float* att1    = sm + o; o += (size_t)CCH*GN*16;
  float* G_kic   = sm + o; o += (size_t)CCH*GN*16;
  float* att2g   = sm + o; o += (size_t)CCH*GN*16;
  float* G_kpt   = sm + o; o += (size_t)PN*289;
  float* S_kpt   = sm + o; o += (size_t)PN*KCH;
  float* G_ikc   = sm + o; o += (size_t)PN*289;
  float* S_ikc   = sm + o; o += (size_t)PN*KCH;
  float* sum_in  = sm + o; o += (size_t)PN*CCH;
  float* sum_ir1 = sm + o; o += (size_t)PN*CCH;
  float* sum_kr  = sm + o; o += (size_t)PN*KCH;
  float* M2      = sm + o; o += (size_t)PN*32*MPAD;  // padded fold matrices
  float* c2      = sm + o; o += (size_t)PN*32;
  float* M1      = sm + o; o += (size_t)PN*32*MPAD;
  float* c1      = sm + o; o += (size_t)PN*32;
  float* s1v     = sm + o; o += (size_t)PN*CK;
  float* s2v     = sm + o; o += (size_t)PN*CK;
  float* ffsum   = sm + o; o += (size_t)PN*CK;
  float* ffmax   = sm + o; o += (size_t)PN*CK;
  float* caS     = sm + o; o += (size_t)PN*CK;
  float* ikcWp   = sm + o; o += (size_t)32*WPAD;     // padded conv weights
  float* ikcBp   = sm + o; o += 32;
  float* kicWp   = sm + o; o += (size_t)32*WPAD;
  float* kicBp   = sm + o; o += 32;
  float* ffWp    = sm + o; o += (size_t)32*WPAD;
  float* ffBp    = sm + o; o += 32;

  // zero all small accumulators / padded matrices each call (graph-safe)
  hipMemsetAsync(sm, 0, o*sizeof(float), stream);

  // Pad conv weights into [32][WPAD]
  k_pad_w<<<7, 256, 0, stream>>>(ikc_w, ikc_b, KCH, ikcWp, ikcBp);
  k_pad_w<<<7, 256, 0, stream>>>(kic_w, kic_b, CCH, kicWp, kicBp);
  k_pad_w<<<7, 256, 0, stream>>>(ff_w,  ff_b,  CCH, ffWp,  ffBp);

  // Phase A: input grams + attention folding
  k_gram_group<<<dim3(CCH*GN*3), 256, 0, stream>>>(inst_f, G_inst, sum_in);
  k_gram_self <<<dim3(PN, HWD/768), 256, 0, stream>>>(kpt_f, G_kpt, S_kpt);
  k_att_group <<<1, 256, 0, stream>>>(G_inst, param, att1);
  k_jr_prep   <<<GN, 320, 0, stream>>>(G_kpt, S_kpt, jr2_w, jr2_b, M2, c2);

  // Phase B: ir1 and kr (+ channel sums for SE)
  k_fuse_ir<<<dim3(CCH*GN, HWD/2048), 256, 0, stream>>>(
      inst_f, att1, ir1, CCH, 0, sum_ir1, nullptr, CCH, 0);
  k_fuse_kr<<<dim3(PN, HWD/128), 256, 0, stream>>>(
      kpt_f, M2, c2, kr, KCH, 0, sum_kr, nullptr, KCH, 0);

  // Phase C: SE scales + WMMA conv1x1 projections
  k_se<<<PN, 64, 0, stream>>>(sum_ir1, CCH, S_kpt, KCH,
                              se1_w1, se1_b1, se1_w2, se1_b2, s1v);
  k_se<<<PN, 64, 0, stream>>>(sum_kr, KCH, sum_in, CCH,
                              se2_w1, se2_b1, se2_w2, se2_b2, s2v);
  k_conv<<<dim3(PN, HWD/128), 256, 0, stream>>>(ir1, CCH, kpt_f, KCH, s1v,
                                                ikcWp, ikcBp, KCH, ikc);
  k_conv<<<dim3(PN, HWD/128), 256, 0, stream>>>(kr, KCH, inst_f, CCH, s2v,
                                                kicWp, kicBp, CCH, kic);

  // Phase D: second-round grams + attention folding
  k_gram_self <<<dim3(PN, HWD/768), 256, 0, stream>>>(ikc, G_ikc, S_ikc);
  k_gram_group<<<dim3(CCH*GN*3), 256, 0, stream>>>(kic, G_kic, nullptr);
  k_jr_prep   <<<GN, 320, 0, stream>>>(G_ikc, S_ikc, jr1_w, jr1_b, M1, c1);
  k_att_group <<<1, 256, 0, stream>>>(G_kic, param, att2g);

  // Phase E: assemble ff = concat(kr1, ir2) directly (aliases dead ir1/kr) + stats
  k_fuse_kr<<<dim3(PN, HWD/128), 256, 0, stream>>>(
      ikc, M1, c1, ff, CK, 0, ffsum, ffmax, CK, 0);
  k_fuse_ir<<<dim3(CCH*GN, HWD/2048), 256, 0, stream>>>(
      kic, att2g, ff, CK, KCH, ffsum, ffmax, CK, KCH);

  // Phase F: channel attention, spatial map, final WMMA conv + residual
  k_chatt<<<PN, 64, 0, stream>>>(ffsum, ffmax, ca_w1, ca_w2, caS);
  k_smap <<<dim3(PN, HWD/256), 256, 0, stream>>>(ff, caS, smap);
  k_final<<<dim3(PN, HWD/128), 256, 0, stream>>>(ff, caS, smap, sa_w,
                                                 ffWp, ffBp, inst_f, (float*)d_out);
}